// PointGroup_47210280517893
// MI455X (gfx1250) — compile-verified
//
#include <hip/hip_runtime.h>
#include <hip/hip_bf16.h>
#include <float.h>

#define NB 2
#define NP 65536
#define NC 13
#define NSEED 10
#define MINPTS 50
#define PTHRESH 0.05f

typedef __attribute__((ext_vector_type(2))) float v2f;
typedef __attribute__((ext_vector_type(4))) float v4f;
typedef __attribute__((ext_vector_type(8))) float v8f;

// ---------------------------------------------------------------- kernel 0
__global__ void pg_init(int* __restrict__ counts, int* __restrict__ firsts) {
    int t = threadIdx.x;
    if (t < NB * NC) { counts[t] = 0; firsts[t] = 0x7fffffff; }
}

// ---------------------------------------------------------------- kernel 1
// Per point: softmax over 13 classes, shifted coords (SoA), per-class mind
// init (+FLT_MAX valid / -FLT_MAX invalid), wave-aggregated count + first.
__global__ void pg_prep(const float* __restrict__ logits,
                        const float* __restrict__ pts,
                        const float* __restrict__ offs,
                        float* __restrict__ sx, float* __restrict__ sy,
                        float* __restrict__ sz,
                        float* __restrict__ mind,
                        int* __restrict__ counts, int* __restrict__ firsts) {
    int i = blockIdx.x * 256 + threadIdx.x;   // 0 .. NB*NP-1
    int b = i >> 16;
    int n = i & (NP - 1);
    float x = pts[3 * i + 0] + offs[3 * i + 0];
    float y = pts[3 * i + 1] + offs[3 * i + 1];
    float z = pts[3 * i + 2] + offs[3 * i + 2];
    sx[i] = x; sy[i] = y; sz[i] = z;

    float lg[NC];
    const float* lp = logits + (size_t)i * NC;
    float mx = -FLT_MAX;
#pragma unroll
    for (int c = 0; c < NC; c++) { lg[c] = lp[c]; mx = fmaxf(mx, lg[c]); }
    float s = 0.f;
#pragma unroll
    for (int c = 0; c < NC; c++) { lg[c] = __expf(lg[c] - mx); s += lg[c]; }
    float inv = __builtin_amdgcn_rcpf(s);

    int lane = threadIdx.x & 31;
#pragma unroll
    for (int c = 0; c < NC; c++) {
        float p = lg[c] * inv;
        bool v = p > PTHRESH;
        mind[((b * NC) + c) * NP + n] = v ? FLT_MAX : -FLT_MAX;
        unsigned m32 = (unsigned)__ballot(v);
        if (lane == 0 && m32) {
            atomicAdd(&counts[b * NC + c], __popc(m32));
            atomicMin(&firsts[b * NC + c], (n & ~31) + __ffs(m32) - 1);
        }
    }
}

// ---------------------------------------------------------------- kernel 2
// FPS: one block (1024 threads) per (batch, class). 9 fused min-update +
// argmax sweeps. JAX tie-break: max value, then minimum index.
__global__ void pg_fps(const float* __restrict__ sx, const float* __restrict__ sy,
                       const float* __restrict__ sz,
                       float* __restrict__ mind,
                       const int* __restrict__ counts, const int* __restrict__ firsts,
                       int* __restrict__ nseeds,
                       float* __restrict__ seedx, float* __restrict__ seedy,
                       float* __restrict__ seedz) {
    int bc = blockIdx.x;
    int b = bc / NC;
    int tid = threadIdx.x;
    __shared__ float rv[1024];
    __shared__ int   ri[1024];
    __shared__ float qx, qy, qz;

    const float* X = sx + b * NP;
    const float* Y = sy + b * NP;
    const float* Z = sz + b * NP;
    float* M = mind + bc * NP;

    if (tid == 0) {
        int cnt = counts[bc];
        int ns = cnt / MINPTS;
        if (ns > NSEED) ns = NSEED;
        if (ns > cnt) ns = cnt;
        nseeds[bc] = ns;
        int i0 = firsts[bc];
        if (i0 == 0x7fffffff) i0 = 0;   // argmax of all-false == 0
        qx = X[i0]; qy = Y[i0]; qz = Z[i0];
        seedx[bc * 16 + 0] = qx; seedy[bc * 16 + 0] = qy; seedz[bc * 16 + 0] = qz;
    }
    __syncthreads();

    for (int r = 1; r < NSEED; r++) {
        float cx = qx, cy = qy, cz = qz;
        float bv = -INFINITY;
        int bi = 0;
        for (int n = tid; n < NP; n += 1024) {
            float mv = M[n];
            float dx = X[n] - cx, dy = Y[n] - cy, dz = Z[n] - cz;
            float d2 = dx * dx + dy * dy + dz * dz;
            float nm = (mv >= 0.f) ? fminf(mv, d2) : mv;   // invalid stays -FLT_MAX
            M[n] = nm;
            if (nm > bv) { bv = nm; bi = n; }              // strict > keeps lowest n
        }
        rv[tid] = bv; ri[tid] = bi;
        __syncthreads();
        for (int s = 512; s > 0; s >>= 1) {
            if (tid < s) {
                float v2 = rv[tid + s]; int i2 = ri[tid + s];
                if (v2 > rv[tid] || (v2 == rv[tid] && i2 < ri[tid])) {
                    rv[tid] = v2; ri[tid] = i2;
                }
            }
            __syncthreads();
        }
        if (tid == 0) {
            int ni = ri[0];
            qx = X[ni]; qy = Y[ni]; qz = Z[ni];
            seedx[bc * 16 + r] = qx; seedy[bc * 16 + r] = qy; seedz[bc * 16 + r] = qz;
        }
        __syncthreads();
    }
    if (tid < 6) {   // zero pad seed slots 10..15 (B-matrix columns)
        seedx[bc * 16 + 10 + tid] = 0.f;
        seedy[bc * 16 + 10 + tid] = 0.f;
        seedz[bc * 16 + 10 + tid] = 0.f;
    }
}

// ---------------------------------------------------------------- kernel 3
// One wave32 per 16-point tile. Two passes over classes (pass loop kept as a
// single code copy to bound register pressure):
//   pass 0: per-class V_WMMA_F32_16X16X4_F32 distances -> Gaussian soft
//           assignment -> accumulate instance-softmax denominator per row.
//   pass 1: recompute, multiply by 1/denominator, coalesced b128 stores.
// Branchless masking (EXEC is never zero for partial seed sets) and
// stage-batched shuffle reductions for cross-row ILP on the ds pipe.
__global__ void __launch_bounds__(256) pg_main(
    const float* __restrict__ logits,
    const float* __restrict__ sx, const float* __restrict__ sy,
    const float* __restrict__ sz,
    const float* __restrict__ seedx, const float* __restrict__ seedy,
    const float* __restrict__ seedz,
    const int* __restrict__ nseeds,
    float* __restrict__ out) {
    const int lane  = threadIdx.x & 31;
    const int wv    = threadIdx.x >> 5;
    const int t     = blockIdx.x * 8 + wv;     // tile id, NB*NP/16 total
    const int b     = t >> 12;
    const int pbase = (t & 4095) << 4;
    const int k     = lane & 15;               // seed slot / column N
    const int hi    = lane >> 4;               // half of the wave
    const int srcM  = 8 * hi;                  // row-half base for bpermute
    const int gi    = b * NP + pbase + k;      // this lane's point

    float x = sx[gi], y = sy[gi], z = sz[gi];
    // A 16x4 f32 layout: lo lanes hold (K=0,K=1) = (x,y); hi lanes (K=2,K=3) = (z,0)
    v2f A;
    A.x = hi ? z : x;
    A.y = hi ? 0.f : y;
    float pn_own = x * x + y * y + z * z;

    // own-point class softmax (kept in registers across both passes)
    float lg[NC];
    const float* lp = logits + (size_t)gi * NC;
    float mx = -FLT_MAX;
#pragma unroll
    for (int c = 0; c < NC; c++) { lg[c] = lp[c]; mx = fmaxf(mx, lg[c]); }
    float ssum = 0.f;
#pragma unroll
    for (int c = 0; c < NC; c++) { lg[c] = __expf(lg[c] - mx); ssum += lg[c]; }
    float sinv = __builtin_amdgcn_rcpf(ssum);

    // D element (vgpr j, lane): row M = j + 8*hi -> |p|^2 of point M
    float pnM[8];
#pragma unroll
    for (int j = 0; j < 8; j++) pnM[j] = __shfl(pn_own, srcM + j, 32);

    float acc[8];   // pass 0: denominator partials; after pass 0: 1/denominator
#pragma unroll
    for (int j = 0; j < 8; j++) acc[j] = 0.f;

    const float inv2s2 = 1.0f / (2.0f * 0.15f * 0.15f);   // sigma = 0.3*0.5

#pragma unroll 1
    for (int pass = 0; pass < 2; ++pass) {
#pragma unroll
        for (int c = 0; c < NC; c++) {
            const int bc = b * NC + c;
            const int ns = nseeds[bc];
            float sxx = seedx[bc * 16 + k];
            float syy = seedy[bc * 16 + k];
            float szz = seedz[bc * 16 + k];
            // B 4x16 f32: lo lanes hold (K=0,K=1) of column k; hi lanes (K=2,K=3)
            v2f Bv;
            Bv.x = hi ? szz : sxx;
            Bv.y = hi ? 0.f : syy;
            float sn = sxx * sxx + syy * syy + szz * szz;
            v8f D = {};
            D = __builtin_amdgcn_wmma_f32_16x16x4_f32(false, A, false, Bv,
                                                      (short)0, D, false, false);
            float svm = (k < ns) ? 1.f : 0.f;   // branchless seed-valid mask
            float p = lg[c] * sinv;
            float pg_own = (p > PTHRESH) ? p : 0.f;

            // Gaussian weights: d2 >= 0 always (pad seeds are zeroed), so the
            // unconditional exp is safe (arg <= 0).
            float w8[8];
#pragma unroll
            for (int j = 0; j < 8; j++) {
                float d2 = pnM[j] + sn - 2.0f * D[j];
                w8[j] = svm * __expf(-d2 * inv2s2);
            }
            // seed-axis sums within each 16-lane half, batched by stage so the
            // 8 independent bpermute chains overlap on the LDS pipe
            float rs8[8];
#pragma unroll
            for (int j = 0; j < 8; j++) rs8[j] = w8[j];
#pragma unroll
            for (int j = 0; j < 8; j++) rs8[j] += __shfl_xor(rs8[j], 1, 16);
#pragma unroll
            for (int j = 0; j < 8; j++) rs8[j] += __shfl_xor(rs8[j], 2, 16);
#pragma unroll
            for (int j = 0; j < 8; j++) rs8[j] += __shfl_xor(rs8[j], 4, 16);
#pragma unroll
            for (int j = 0; j < 8; j++) rs8[j] += __shfl_xor(rs8[j], 8, 16);
            float pgM8[8];
#pragma unroll
            for (int j = 0; j < 8; j++) pgM8[j] = __shfl(pg_own, srcM + j, 32);

            float e8[8];
#pragma unroll
            for (int j = 0; j < 8; j++) {
                float fin = w8[j] * __builtin_amdgcn_rcpf(rs8[j] + 1e-8f) * pgM8[j];
                e8[j] = svm * __expf(10.0f * fin);   // logits in [0,10]: stable
            }
            if (pass == 0) {
#pragma unroll
                for (int j = 0; j < 8; j++) acc[j] += e8[j];  // lane-local over classes
            } else if (k < NSEED) {
                // lane owns 8 consecutive points of output row (c*10+k)
                float* op = out + ((size_t)(b * (NC * NSEED) + c * NSEED + k)) * NP
                            + pbase + srcM;
                v4f lo4 = { e8[0] * acc[0], e8[1] * acc[1], e8[2] * acc[2], e8[3] * acc[3] };
                v4f hi4 = { e8[4] * acc[4], e8[5] * acc[5], e8[6] * acc[6], e8[7] * acc[7] };
                *(v4f*)(op)     = lo4;
                *(v4f*)(op + 4) = hi4;
            }
        }
        if (pass == 0) {
            // reduce denominators over the 16-lane halves (stage-batched), invert
#pragma unroll
            for (int j = 0; j < 8; j++) acc[j] += __shfl_xor(acc[j], 1, 16);
#pragma unroll
            for (int j = 0; j < 8; j++) acc[j] += __shfl_xor(acc[j], 2, 16);
#pragma unroll
            for (int j = 0; j < 8; j++) acc[j] += __shfl_xor(acc[j], 4, 16);
#pragma unroll
            for (int j = 0; j < 8; j++) acc[j] += __shfl_xor(acc[j], 8, 16);
#pragma unroll
            for (int j = 0; j < 8; j++)
                acc[j] = (acc[j] > 0.f) ? __builtin_amdgcn_rcpf(acc[j]) : 0.f;
        }
    }
}

// ---------------------------------------------------------------- launch
extern "C" void kernel_launch(void* const* d_in, const int* in_sizes, int n_in,
                              void* d_out, int out_size, void* d_ws, size_t ws_size,
                              hipStream_t stream) {
    (void)in_sizes; (void)n_in; (void)out_size; (void)ws_size;
    const float* logits = (const float*)d_in[0];
    const float* pts    = (const float*)d_in[1];
    const float* offs   = (const float*)d_in[2];
    float* out = (float*)d_out;

    char* w = (char*)d_ws;
    size_t o = 0;
    auto alloc = [&](size_t bytes) -> void* {
        void* p = w + o;
        o += (bytes + 255) & ~(size_t)255;
        return p;
    };
    float* sx    = (float*)alloc((size_t)NB * NP * 4);
    float* sy    = (float*)alloc((size_t)NB * NP * 4);
    float* sz    = (float*)alloc((size_t)NB * NP * 4);
    float* mind  = (float*)alloc((size_t)NB * NC * NP * 4);
    int*   counts= (int*)  alloc((size_t)NB * NC * 4);
    int*   firsts= (int*)  alloc((size_t)NB * NC * 4);
    int*   nseeds= (int*)  alloc((size_t)NB * NC * 4);
    float* seedx = (float*)alloc((size_t)NB * NC * 16 * 4);
    float* seedy = (float*)alloc((size_t)NB * NC * 16 * 4);
    float* seedz = (float*)alloc((size_t)NB * NC * 16 * 4);

    pg_init<<<1, 32, 0, stream>>>(counts, firsts);
    pg_prep<<<(NB * NP) / 256, 256, 0, stream>>>(logits, pts, offs,
                                                 sx, sy, sz, mind, counts, firsts);
    pg_fps<<<NB * NC, 1024, 0, stream>>>(sx, sy, sz, mind, counts, firsts,
                                         nseeds, seedx, seedy, seedz);
    pg_main<<<(NB * NP / 16) / 8, 256, 0, stream>>>(logits, sx, sy, sz,
                                                    seedx, seedy, seedz, nseeds, out);
}